// Net_KAN_60859686584588
// MI455X (gfx1250) — compile-verified
//
#include <hip/hip_runtime.h>
#include <hip/hip_bf16.h>

typedef __attribute__((ext_vector_type(2))) float v2f;
typedef __attribute__((ext_vector_type(8))) float v8f;

#define NN      50000
#define NE      800000
#define F       64
#define NG      64
#define ODIM    3

// ---------------- degree / normalization ----------------

__global__ void k_deg_init(float* deg) {
    int i = blockIdx.x * blockDim.x + threadIdx.x;
    if (i < NN) deg[i] = 1.0f;   // self-loop weight
}

__global__ void k_deg_accum(const int* __restrict__ ei, const float* __restrict__ w,
                            float* __restrict__ deg) {
    int e = blockIdx.x * blockDim.x + threadIdx.x;
    if (e < NE) atomicAdd(&deg[ei[NE + e]], w[e]);   // dst-side degree
}

__global__ void k_deg_finalize(float* deg) {
    int i = blockIdx.x * blockDim.x + threadIdx.x;
    if (i < NN) {
        float d = deg[i];                       // >= 1 always (self-loop)
        deg[i] = rsqrtf(fmaxf(d, 1e-12f));      // becomes dis[]
    }
}

// ---------------- WMMA f32 GEMM: H[NNx64] = act(A[NNx64]) @ W[64x64] ----------------
// One wave computes a 16x16 tile; blockDim=(32,4): 4 waves cover the 4 N-tiles.

__global__ __launch_bounds__(128)
void k_gemm_wmma(const float* __restrict__ A, const float* __restrict__ W,
                 float* __restrict__ H, int relu_in) {
    const int lane = threadIdx.x;            // 0..31
    const int n0   = threadIdx.y << 4;       // 0,16,32,48
    const int m0   = blockIdx.x << 4;
    const int m    = lane & 15;
    const int kb   = (lane >> 4) << 1;       // 0 (lanes 0-15) or 2 (lanes 16-31)
    const int nn   = n0 + (lane & 15);

    const float* arow = A + (size_t)(m0 + m) * F;
    v8f c = {};
    #pragma unroll
    for (int k0 = 0; k0 < F; k0 += 4) {
        float a0 = arow[k0 + kb];
        float a1 = arow[k0 + kb + 1];
        if (relu_in) { a0 = fmaxf(a0, 0.0f); a1 = fmaxf(a1, 0.0f); }
        v2f a; a.x = a0; a.y = a1;
        v2f b; b.x = W[(k0 + kb) * F + nn];
               b.y = W[(k0 + kb + 1) * F + nn];
        // (neg_a, A, neg_b, B, c_mod, C, reuse_a, reuse_b)
        c = __builtin_amdgcn_wmma_f32_16x16x4_f32(false, a, false, b,
                                                  (short)0, c, false, false);
    }
    // C/D layout: vgpr r -> row m0 + r + (lane>=16 ? 8 : 0), col = n0 + (lane&15)
    float* hrow = H + (size_t)(m0 + ((lane >> 4) << 3)) * F + nn;
    #pragma unroll
    for (int r = 0; r < 8; ++r)
        hrow[(size_t)r * F] = c[r];
}

// ---------------- aggregation ----------------
// agg[i][f] = dis[i]^2 * H[i][f] + bias[f]   (self-loop term, norm = dis*1*dis)

__global__ void k_agg_init(const float* __restrict__ H, const float* __restrict__ dis,
                           const float* __restrict__ bias, float* __restrict__ agg) {
    int gid = blockIdx.x * blockDim.x + threadIdx.x;   // NN*F threads
    if (gid < NN * F) {
        int node = gid >> 6;
        int f    = gid & 63;
        float d  = dis[node];
        agg[gid] = d * d * H[gid] + bias[f];
    }
}

// agg[dst] += dis[src]*w*dis[dst] * H[src]; 16 threads per edge, float4 each
__global__ void k_agg_edges(const int* __restrict__ ei, const float* __restrict__ w,
                            const float* __restrict__ dis, const float* __restrict__ H,
                            float* __restrict__ agg) {
    unsigned gid = blockIdx.x * blockDim.x + threadIdx.x;
    unsigned e = gid >> 4;
    if (e >= NE) return;
    int q = (int)(gid & 15);
    int s = ei[e];
    int d = ei[NE + e];
    float coeff = dis[s] * w[e] * dis[d];
    const float4 hv = *(const float4*)(H + (size_t)s * F + q * 4);
    float* dp = agg + (size_t)d * F + q * 4;
    atomicAdd(dp + 0, coeff * hv.x);
    atomicAdd(dp + 1, coeff * hv.y);
    atomicAdd(dp + 2, coeff * hv.z);
    atomicAdd(dp + 3, coeff * hv.w);
}

// ---------------- pooling ----------------

__global__ void k_zero_pool(float* sums, float* cnt) {
    int i = blockIdx.x * blockDim.x + threadIdx.x;
    if (i < NG * F) sums[i] = 0.0f;
    if (i < NG)     cnt[i]  = 0.0f;
}

__global__ void k_pool_accum(const float* __restrict__ h, const int* __restrict__ batch,
                             float* __restrict__ sums, float* __restrict__ cnt) {
    unsigned gid = blockIdx.x * blockDim.x + threadIdx.x;   // NN*16 threads
    unsigned node = gid >> 4;
    if (node >= NN) return;
    int q = (int)(gid & 15);
    int g = batch[node];
    const float4 hv = *(const float4*)(h + (size_t)node * F + q * 4);
    float* sp = sums + (size_t)g * F + q * 4;
    atomicAdd(sp + 0, hv.x);
    atomicAdd(sp + 1, hv.y);
    atomicAdd(sp + 2, hv.z);
    atomicAdd(sp + 3, hv.w);
    if (q == 0) atomicAdd(&cnt[g], 1.0f);
}

// ---------------- KAN head ----------------
// hbuf[g][0..63] = relu(pooled@kw1+kb1), hbuf[g][64..127] = relu(pooled@kw2+kb2)

__global__ __launch_bounds__(128)
void k_head_hidden(const float* __restrict__ sums, const float* __restrict__ cnt,
                   const float* __restrict__ kw1, const float* __restrict__ kb1,
                   const float* __restrict__ kw2, const float* __restrict__ kb2,
                   float* __restrict__ hbuf) {
    int g = blockIdx.x;
    int j = threadIdx.x;                 // 0..127
    const float* kw = (j < 64) ? kw1 : kw2;
    const float* kb = (j < 64) ? kb1 : kb2;
    int jj = j & 63;
    float inv = 1.0f / fmaxf(cnt[g], 1.0f);
    float acc = kb[jj];
    #pragma unroll 8
    for (int k = 0; k < F; ++k)
        acc = fmaf(sums[g * F + k] * inv, kw[k * F + jj], acc);
    hbuf[g * 128 + j] = fmaxf(acc, 0.0f);
}

__global__ __launch_bounds__(192)
void k_head_out(const float* __restrict__ hbuf, const float* __restrict__ kwo,
                const float* __restrict__ kbo, float* __restrict__ out) {
    int t = threadIdx.x;                 // 0..191
    int g = t / ODIM;
    int o = t % ODIM;
    float acc = kbo[o];
    #pragma unroll 8
    for (int j = 0; j < 2 * F; ++j)
        acc = fmaf(hbuf[g * 128 + j], kwo[j * ODIM + o], acc);
    out[g * ODIM + o] = acc;
}

// ---------------- launcher ----------------

extern "C" void kernel_launch(void* const* d_in, const int* in_sizes, int n_in,
                              void* d_out, int out_size, void* d_ws, size_t ws_size,
                              hipStream_t stream) {
    const float* x    = (const float*)d_in[0];
    const int*   ei   = (const int*)  d_in[1];   // [2, NE] flat: src row then dst row
    const float* ew   = (const float*)d_in[2];
    const int*   bat  = (const int*)  d_in[3];
    const float* W1   = (const float*)d_in[4];
    const float* b1   = (const float*)d_in[5];
    const float* W2   = (const float*)d_in[6];
    const float* b2   = (const float*)d_in[7];
    const float* kw1  = (const float*)d_in[8];
    const float* kb1  = (const float*)d_in[9];
    const float* kw2  = (const float*)d_in[10];
    const float* kb2  = (const float*)d_in[11];
    const float* kwo  = (const float*)d_in[12];
    const float* kbo  = (const float*)d_in[13];
    float* out = (float*)d_out;

    // workspace carve-up (bytes), 256B aligned
    char* ws = (char*)d_ws;
    size_t off = 0;
    float* dis  = (float*)(ws + off); off += ((size_t)NN * 4 + 255) & ~(size_t)255;
    float* bufH = (float*)(ws + off); off += (size_t)NN * F * 4;           // GEMM out
    float* bufA = (float*)(ws + off); off += (size_t)NN * F * 4;           // aggregated
    float* sums = (float*)(ws + off); off += (size_t)NG * F * 4;
    float* cnt  = (float*)(ws + off); off += 256;
    float* hbuf = (float*)(ws + off); off += (size_t)NG * 128 * 4;

    const int B = 256;
    dim3 gemmBlk(32, 4);

    // --- normalization coefficients ---
    k_deg_init     <<<(NN + B - 1) / B, B, 0, stream>>>(dis);
    k_deg_accum    <<<(NE + B - 1) / B, B, 0, stream>>>(ei, ew, dis);
    k_deg_finalize <<<(NN + B - 1) / B, B, 0, stream>>>(dis);

    // --- layer 1: H = X @ W1 ; agg1 = propagate(H) + b1 ---
    k_gemm_wmma <<<NN / 16, gemmBlk, 0, stream>>>(x, W1, bufH, 0);
    k_agg_init  <<<(NN * F + B - 1) / B, B, 0, stream>>>(bufH, dis, b1, bufA);
    k_agg_edges <<<(NE * 16 + B - 1) / B, B, 0, stream>>>(ei, ew, dis, bufH, bufA);

    // --- layer 2: H = relu(agg1) @ W2 ; agg2 = propagate(H) + b2 ---
    k_gemm_wmma <<<NN / 16, gemmBlk, 0, stream>>>(bufA, W2, bufH, 1);
    k_agg_init  <<<(NN * F + B - 1) / B, B, 0, stream>>>(bufH, dis, b2, bufA);
    k_agg_edges <<<(NE * 16 + B - 1) / B, B, 0, stream>>>(ei, ew, dis, bufH, bufA);

    // --- global mean pool ---
    k_zero_pool  <<<(NG * F + B - 1) / B, B, 0, stream>>>(sums, cnt);
    k_pool_accum <<<(NN * 16 + B - 1) / B, B, 0, stream>>>(bufA, bat, sums, cnt);

    // --- KAN head ---
    k_head_hidden <<<NG, 128, 0, stream>>>(sums, cnt, kw1, kb1, kw2, kb2, hbuf);
    k_head_out    <<<1, 192, 0, stream>>>(hbuf, kwo, kbo, out);
}